// SparseMaxNorm_31748398252185
// MI455X (gfx1250) — compile-verified
//
#include <hip/hip_runtime.h>

// SparseMaxNorm for MI455X (gfx1250, wave32).
// Memory-bound scatter-max + gather-normalize; no matrix ops apply.
// Strategy: 128-bit vector memory ops for all streams, u32-bit-pattern
// atomic max (exact for non-negative floats) hitting the 192MB L2,
// non-temporal stores for the write-only output stream.

typedef float  v4f __attribute__((ext_vector_type(4)));
typedef int    v4i __attribute__((ext_vector_type(4)));

#define EPSILON 1e-5f

// ---------------------------------------------------------------------------
// Kernel 1: new_max (in d_out tail) = max_x.  Re-run every call so the
// subsequent atomics are deterministic across graph replays.
// ---------------------------------------------------------------------------
__global__ __launch_bounds__(256)
void SMN_init_max(const float* __restrict__ max_x,
                  float* __restrict__ new_max,
                  int n) {
    int i = (blockIdx.x * 256 + threadIdx.x) * 4;
    if (i + 3 < n) {
        v4f v = *reinterpret_cast<const v4f*>(max_x + i);
        *reinterpret_cast<v4f*>(new_max + i) = v;
    } else {
        for (int k = i; k < n; ++k) new_max[k] = max_x[k];
    }
}

// ---------------------------------------------------------------------------
// Kernel 2: scatter-max of |values| into new_max.
// new_max >= 0 and |v| >= 0, so the IEEE-754 bit pattern of a non-negative
// float is monotone in its value -> unsigned-int atomicMax is exact and
// lowers to one global_atomic_max_u32 (L2-resident table, no CAS loop).
// ---------------------------------------------------------------------------
__global__ __launch_bounds__(256)
void SMN_scatter_max(const float* __restrict__ values,
                     const int*   __restrict__ indices,
                     unsigned int* __restrict__ new_max_u,
                     int nnz) {
    int i = (blockIdx.x * 256 + threadIdx.x) * 4;
    if (i + 3 < nnz) {
        v4f v   = *reinterpret_cast<const v4f*>(values  + i);
        v4i idx = *reinterpret_cast<const v4i*>(indices + i);
        atomicMax(new_max_u + idx.x, __float_as_uint(fabsf(v.x)));
        atomicMax(new_max_u + idx.y, __float_as_uint(fabsf(v.y)));
        atomicMax(new_max_u + idx.z, __float_as_uint(fabsf(v.z)));
        atomicMax(new_max_u + idx.w, __float_as_uint(fabsf(v.w)));
    } else {
        for (int k = i; k < nnz; ++k) {
            atomicMax(new_max_u + indices[k], __float_as_uint(fabsf(values[k])));
        }
    }
}

// ---------------------------------------------------------------------------
// Kernel 3: gather updated maxima (L2 hits), normalize+clip, add gathered
// bias (L2 hits), stream result out with non-temporal b128 stores.
// ---------------------------------------------------------------------------
__device__ __forceinline__ float smn_one(float v, int idx,
                                         const float* __restrict__ new_max,
                                         const float* __restrict__ bias) {
    float m = new_max[idx];
    float d = fmaxf(m, EPSILON);
    float n = v / d;
    n = fminf(fmaxf(n, -1.0f), 1.0f);
    return n + bias[idx];
}

__global__ __launch_bounds__(256)
void SMN_gather_norm(const float* __restrict__ values,
                     const int*   __restrict__ indices,
                     const float* __restrict__ new_max,
                     const float* __restrict__ bias,
                     float* __restrict__ out,
                     int nnz) {
    int i = (blockIdx.x * 256 + threadIdx.x) * 4;
    if (i + 3 < nnz) {
        v4f v   = *reinterpret_cast<const v4f*>(values  + i);
        v4i idx = *reinterpret_cast<const v4i*>(indices + i);
        v4f o;
        o.x = smn_one(v.x, idx.x, new_max, bias);
        o.y = smn_one(v.y, idx.y, new_max, bias);
        o.z = smn_one(v.z, idx.z, new_max, bias);
        o.w = smn_one(v.w, idx.w, new_max, bias);
        // write-only 64MB stream: keep it out of L2 so the L2-resident
        // value/index streams and max/bias tables survive.
        __builtin_nontemporal_store(o, reinterpret_cast<v4f*>(out + i));
    } else {
        for (int k = i; k < nnz; ++k) {
            out[k] = smn_one(values[k], indices[k], new_max, bias);
        }
    }
}

// ---------------------------------------------------------------------------
// Launch: inputs in setup_inputs() order:
//   d_in[0] = values_x  [nnz]        float32
//   d_in[1] = max_x     [input_size] float32
//   d_in[2] = bias_x    [input_size] float32
//   d_in[3] = indices_x [nnz]        int
// d_out = out_values [nnz] ++ new_max_x [input_size]
// ---------------------------------------------------------------------------
extern "C" void kernel_launch(void* const* d_in, const int* in_sizes, int n_in,
                              void* d_out, int out_size, void* d_ws, size_t ws_size,
                              hipStream_t stream) {
    const float* values  = (const float*)d_in[0];
    const float* max_x   = (const float*)d_in[1];
    const float* bias    = (const float*)d_in[2];
    const int*   indices = (const int*)d_in[3];

    const int nnz        = in_sizes[0];
    const int input_size = in_sizes[1];

    float* out_values = (float*)d_out;
    float* new_max    = (float*)d_out + nnz;

    const int B = 256;
    const int VEC = 4;

    int init_blocks = (input_size + B * VEC - 1) / (B * VEC);
    SMN_init_max<<<init_blocks, B, 0, stream>>>(max_x, new_max, input_size);

    int nnz_blocks = (nnz + B * VEC - 1) / (B * VEC);
    SMN_scatter_max<<<nnz_blocks, B, 0, stream>>>(
        values, indices, reinterpret_cast<unsigned int*>(new_max), nnz);

    SMN_gather_norm<<<nnz_blocks, B, 0, stream>>>(
        values, indices, new_max, bias, out_values, nnz);
}